// UltraFastBEV_11656541241445
// MI455X (gfx1250) — compile-verified
//
#include <hip/hip_runtime.h>
#include <stdint.h>

// ---------------------------------------------------------------------------
// UltraFastBEV on MI455X (gfx1250, wave32).
// Scatter-histogram workload: no GEMM structure -> WMMA inapplicable.
// CDNA5 paths used: global_load_async_to_lds_b128 (ASYNCcnt) double-buffered
// point staging, f32 global atomics into an L2-resident accumulator,
// non-temporal b128 stores for the write-once zero channels.
// ---------------------------------------------------------------------------

#define BEV_W   256
#define BEV_H   256
#define HW      (BEV_W * BEV_H)   // 65536
#define NF      64
#define NBATCH  4

typedef float f32x4 __attribute__((ext_vector_type(4)));

// Per-lane async stage of one 16-byte point into this lane's LDS slot.
// vdst = LDS byte address, vaddr = 64-bit global address, saddr = off.
__device__ __forceinline__ void async_stage_b128(uint32_t lds_byte_addr, uint64_t gaddr) {
    asm volatile("global_load_async_to_lds_b128 %0, %1, off"
                 :: "v"(lds_byte_addr), "v"(gaddr)
                 : "memory");
}

// ---------------------------------------------------------------------------
// Kernel 1: zero-fill.
//  - accumulator (3 MB): regular stores (hot soon: atomic RMWs want it in L2)
//  - output channels 4..63 per batch (60 MB): non-temporal stores (write-once,
//    never re-read on device -> don't burn L2 capacity on zero lines)
// Channels 0..3 are NOT touched here; bev_finalize writes them.
// ---------------------------------------------------------------------------
__global__ __launch_bounds__(256) void bev_zero(f32x4* __restrict__ out4,
                                                f32x4* __restrict__ acc4, int n_acc4) {
    const int gid    = blockIdx.x * blockDim.x + threadIdx.x;
    const int stride = gridDim.x * blockDim.x;
    const f32x4 z = (f32x4)0.0f;

    for (int i = gid; i < n_acc4; i += stride)
        acc4[i] = z;

    // Per batch: channels 4..63 are a contiguous run of 60*HW floats.
    const int nz4 = (60 * HW) / 4;            // 983040 f32x4 per batch
    #pragma unroll
    for (int b = 0; b < NBATCH; ++b) {
        f32x4* dst = out4 + (size_t)b * ((NF * HW) / 4) + (4 * HW) / 4;
        for (int i = gid; i < nz4; i += stride)
            __builtin_nontemporal_store(z, dst + i);
    }
}

// ---------------------------------------------------------------------------
// Kernel 2: scatter. Persistent grid; each lane double-buffers its own point
// stream through LDS with async loads, then does 3 f32 atomics per valid
// point into acc laid out as [B][3][HW] (coalesced for the finalize reads).
// ---------------------------------------------------------------------------
__global__ __launch_bounds__(256) void bev_scatter(const float* __restrict__ pts,
                                                   float* __restrict__ acc,
                                                   int npts,   // points per batch
                                                   int total)  // NBATCH * npts
{
    __shared__ float4 stage[2][256];
    const int tid    = threadIdx.x;
    const int stride = gridDim.x * 256;
    const uint64_t base = (uint64_t)(uintptr_t)pts;

    const uint32_t lds0 = (uint32_t)(uintptr_t)(&stage[0][tid]);
    const uint32_t lds1 = (uint32_t)(uintptr_t)(&stage[1][tid]);

    int i = blockIdx.x * 256 + tid;

    // Prologue: stage first chunk into buffer 0 (clamped dummy if past end).
    {
        const int ic = (i < total) ? i : 0;
        async_stage_b128(lds0, base + (uint64_t)ic * 16u);
    }

    int buf = 0;
    for (; i < total; i += stride) {
        // Stage next chunk into the other buffer while we consume this one.
        const int inext = i + stride;
        const int ic    = (inext < total) ? inext : 0;
        async_stage_b128(buf ? lds0 : lds1, base + (uint64_t)ic * 16u);

        // Async loads complete in order: <=1 outstanding means the older
        // (current-buffer) copy has landed in LDS.
        asm volatile("s_wait_asynccnt 1" ::: "memory");

        const float4 p = stage[buf][tid];     // ds_load_b128, own slot
        const float x = p.x, y = p.y, zz = p.z, w = p.w;

        // Masked points scatter exact zeros in the reference == no-op here.
        if (x >= -50.0f && x < 50.0f &&
            y >= -50.0f && y < 50.0f &&
            zz >= -3.0f && zz < 5.0f) {
            int xi = (int)((x + 50.0f) * 2.56f);   // 1 / 0.390625
            int yi = (int)((y + 50.0f) * 2.56f);
            xi = xi < 0 ? 0 : (xi > BEV_W - 1 ? BEV_W - 1 : xi);
            yi = yi < 0 ? 0 : (yi > BEV_H - 1 ? BEV_H - 1 : yi);
            const int cell = (yi << 8) + xi;
            const int b    = i / npts;
            float* a = acc + (size_t)b * (3 * HW);
            atomicAdd(a + cell,          1.0f);   // count
            atomicAdd(a + HW + cell,     zz);     // sum z   (feeds ch1 and ch2)
            atomicAdd(a + 2 * HW + cell, w);      // sum intensity
        }
        buf ^= 1;
    }

    // Retire the trailing dummy stage before wave end.
    asm volatile("s_wait_asynccnt 0" ::: "memory");
}

// ---------------------------------------------------------------------------
// Kernel 3: finalize. One thread per (batch, cell); writes channels 0..3.
// ---------------------------------------------------------------------------
__global__ __launch_bounds__(256) void bev_finalize(const float* __restrict__ acc,
                                                    float* __restrict__ out) {
    const int i = blockIdx.x * blockDim.x + threadIdx.x;
    if (i >= NBATCH * HW) return;
    const int b    = i >> 16;         // HW == 65536
    const int cell = i & (HW - 1);

    const float* a = acc + (size_t)b * (3 * HW);
    const float cnt = a[cell];
    const float sz  = a[HW + cell];
    const float si  = a[2 * HW + cell];

    const float ch0 = log1pf(cnt);
    const float div = fmaxf(expf(ch0), 1.0f);   // == max(cnt + 1, 1)

    float* o = out + (size_t)b * (NF * HW) + cell;
    o[0]      = ch0;
    o[HW]     = sz;
    o[2 * HW] = sz / div;
    o[3 * HW] = si / div;
}

// ---------------------------------------------------------------------------
// Host entry.
// ---------------------------------------------------------------------------
extern "C" void kernel_launch(void* const* d_in, const int* in_sizes, int n_in,
                              void* d_out, int out_size, void* d_ws, size_t ws_size,
                              hipStream_t stream) {
    const float* pts = (const float*)d_in[0];
    float* out = (float*)d_out;
    float* acc = (float*)d_ws;                 // NBATCH * 3 * HW floats = 3 MB

    const int total = in_sizes[0] / 4;         // number of points (B*N)
    const int npts  = total / NBATCH;          // points per batch

    const int n_acc4 = (NBATCH * 3 * HW) / 4;  // accumulator as f32x4

    bev_zero<<<2048, 256, 0, stream>>>((f32x4*)out, (f32x4*)acc, n_acc4);
    bev_scatter<<<512, 256, 0, stream>>>(pts, acc, npts, total);

    const int fin = NBATCH * HW;
    bev_finalize<<<(fin + 255) / 256, 256, 0, stream>>>(acc, out);
}